// GCN_22557168239484
// MI455X (gfx1250) — compile-verified
//
#include <hip/hip_runtime.h>
#include <hip/hip_bf16.h>

typedef __attribute__((ext_vector_type(2))) float v2f;
typedef __attribute__((ext_vector_type(4))) float v4f;
typedef __attribute__((ext_vector_type(8))) float v8f;

// ---------------------------------------------------------------------------
// Utility kernels
// ---------------------------------------------------------------------------
__global__ void gcn_fill_f32(float* __restrict__ p, float v, int n) {
    int i = blockIdx.x * blockDim.x + threadIdx.x;
    if (i < n) p[i] = v;
}

// n must be a multiple of 4 (true for all our buffers: N and 16N, N=200000)
__global__ void gcn_fill4_f32(float* __restrict__ p, float v, int n4) {
    int i = blockIdx.x * blockDim.x + threadIdx.x;
    if (i < n4) {
        v4f q = {v, v, v, v};
        *(v4f*)(p + (size_t)i * 4) = q;
    }
}

__global__ void gcn_deg_accum(const int* __restrict__ dst, float* __restrict__ deg, int E) {
    int e = blockIdx.x * blockDim.x + threadIdx.x;
    if (e < E) atomicAdd(&deg[dst[e]], 1.0f);
}

__global__ void gcn_rsqrt(float* __restrict__ d, int N) {
    int i = blockIdx.x * blockDim.x + threadIdx.x;
    if (i < N) d[i] = rsqrtf(d[i]);   // deg >= 1 always (self-loop)
}

// ---------------------------------------------------------------------------
// GEMM: hW[16-row tile][16 cols] = A[16 x 16(padded)] @ Wpad[16 x 16]
// V_WMMA_F32_16X16X4_F32, K=16 as 4 chained WMMAs, full fp32 precision.
// A frag (16x4 f32): lane L: M=L%16; VGPR0=K(2*half), VGPR1=K(2*half+1)
// B frag (4x16):     lane L: N=L%16; same K split  -> ds_load_2addr_b32
// D (16x16):         lane L, VGPR r: D[r + 8*half][L%16]
// Out-of-range rows: clamp the *address* (garbage only pollutes that D row,
// which is never stored) -> unconditional b64 loads, no exec-mask churn.
// ---------------------------------------------------------------------------
__global__ void gcn_gemm_wmma(const float* __restrict__ A, const float* __restrict__ W,
                              float* __restrict__ out, int N, int Fin, int Fout) {
    __shared__ float Wpad[16][16];
    int t = threadIdx.x;
    {   // 256 threads cover the 16x16 padded weight tile exactly
        int r = t >> 4, c = t & 15;
        Wpad[r][c] = (r < Fin && c < Fout) ? W[r * Fout + c] : 0.0f;
    }
    __syncthreads();

    int wave = t >> 5;          // 8 waves per block, one 16-row tile each
    int lane = t & 31;
    int tile = blockIdx.x * 8 + wave;
    int numTiles = (N + 15) >> 4;
    if (tile >= numTiles) return;           // wave-uniform: EXEC stays all-1s

    int m0   = tile << 4;
    int M    = lane & 15;
    int half = lane >> 4;
    int row  = m0 + M;
    int rowc = row < N ? row : N - 1;       // clamp, never predicate the load
    const float* rowp = A + (size_t)rowc * 16 + 2 * half;

    v8f acc = {};
#pragma unroll
    for (int k0 = 0; k0 < 16; k0 += 4) {
        v2f a = *(const v2f*)(rowp + k0);           // one global_load_b64
        v2f b;
        b.x = Wpad[k0 + 2 * half][M];
        b.y = Wpad[k0 + 2 * half + 1][M];           // ds_load_2addr_b32
        acc = __builtin_amdgcn_wmma_f32_16x16x4_f32(
            false, a, false, b, (short)0, acc, false, false);
    }

    int n = lane & 15;
    float* outp = out + (size_t)(m0 + 8 * half) * 16 + n;
    if (m0 + 16 <= N) {                      // fast path: full tile
#pragma unroll
        for (int r = 0; r < 8; ++r) outp[(size_t)r * 16] = acc[r];
    } else {                                 // tail tile (unused for N%16==0)
#pragma unroll
        for (int r = 0; r < 8; ++r)
            if (m0 + r + 8 * half < N) outp[(size_t)r * 16] = acc[r];
    }
}

// ---------------------------------------------------------------------------
// Edge scatter: one thread per (edge, feature). agg[dst][f] += hW[src][f]*norm
// dinv table (800KB) and agg table (12.8MB) are L2-resident on MI455X.
// ---------------------------------------------------------------------------
template <int LOGF>
__global__ void gcn_scatter(const int* __restrict__ src, const int* __restrict__ dst,
                            const float* __restrict__ dinv, const float* __restrict__ hW,
                            float* __restrict__ agg, int E) {
    int idx = blockIdx.x * blockDim.x + threadIdx.x;
    int e = idx >> LOGF;
    if (e >= E) return;
    int f = idx & ((1 << LOGF) - 1);
    int s = src[e];
    int d = dst[e];
    float nrm = dinv[s] * dinv[d];
    atomicAdd(&agg[(size_t)d * 16 + f], hW[(size_t)s * 16 + f] * nrm);
}

// ---------------------------------------------------------------------------
// Epilogue (hidden layers): h = mish(agg + selfloop + bias), zero the padding.
// 4 threads per node, float4 per thread. Fout in {8,16} -> each 4-wide chunk
// is either fully real or fully padding.
// ---------------------------------------------------------------------------
__global__ void gcn_epilogue_mish(const float* __restrict__ agg, const float* __restrict__ hW,
                                  const float* __restrict__ dinv, const float* __restrict__ b,
                                  float* __restrict__ h, int N, int Fout) {
    int idx = blockIdx.x * blockDim.x + threadIdx.x;   // N*4 threads
    if (idx >= N * 4) return;
    int i  = idx >> 2;
    int f0 = (idx & 3) * 4;
    v4f val = {};
    if (f0 < Fout) {
        float di = dinv[i];
        float dd = di * di;
        size_t base = (size_t)i * 16 + f0;
        v4f ag = *(const v4f*)(agg + base);
        v4f hw = *(const v4f*)(hW + base);
#pragma unroll
        for (int j = 0; j < 4; ++j) {
            float v = ag[j] + hw[j] * dd + b[f0 + j];
            float sp = (v > 20.0f) ? v : log1pf(expf(v));   // softplus
            val[j] = v * tanhf(sp);                          // mish
        }
    }
    *(v4f*)(h + (size_t)i * 16 + f0) = val;
}

// ---------------------------------------------------------------------------
// Final layer: v = agg + selfloop + bias (Fout=2), then log_softmax over 2.
// ---------------------------------------------------------------------------
__global__ void gcn_logsoftmax2(const float* __restrict__ agg, const float* __restrict__ hW,
                                const float* __restrict__ dinv, const float* __restrict__ b,
                                float* __restrict__ out, int N) {
    int i = blockIdx.x * blockDim.x + threadIdx.x;
    if (i >= N) return;
    float di = dinv[i];
    float dd = di * di;
    size_t base = (size_t)i * 16;
    v2f ag = *(const v2f*)(agg + base);
    v2f hw = *(const v2f*)(hW + base);
    float v0 = ag.x + hw.x * dd + b[0];
    float v1 = ag.y + hw.y * dd + b[1];
    float m = fmaxf(v0, v1);
    float lse = m + logf(expf(v0 - m) + expf(v1 - m));
    v2f o = {v0 - lse, v1 - lse};
    *(v2f*)(out + (size_t)i * 2) = o;
}

// ---------------------------------------------------------------------------
// Host-side launch (graph-capture safe: only kernel launches on `stream`)
// ---------------------------------------------------------------------------
extern "C" void kernel_launch(void* const* d_in, const int* in_sizes, int n_in,
                              void* d_out, int out_size, void* d_ws, size_t ws_size,
                              hipStream_t stream) {
    (void)n_in; (void)out_size; (void)ws_size;

    const float* x   = (const float*)d_in[0];          // [N,16]
    const int*  edge = (const int*)  d_in[1];          // [2,E] row-major
    const float* W1  = (const float*)d_in[2];
    const float* b1  = (const float*)d_in[3];
    const float* W2  = (const float*)d_in[4];
    const float* b2  = (const float*)d_in[5];
    const float* W3  = (const float*)d_in[6];
    const float* b3  = (const float*)d_in[7];
    const float* W4  = (const float*)d_in[8];
    const float* b4  = (const float*)d_in[9];
    float* out = (float*)d_out;

    const int N = in_sizes[0] / 16;
    const int E = in_sizes[1] / 2;
    const int* src = edge;          // edge_index[0]
    const int* dst = edge + E;      // edge_index[1]

    // Workspace layout (floats): dinv[N] | h[16N] | hW[16N] | agg[16N]  (~39MB)
    float* ws   = (float*)d_ws;
    float* dinv = ws;
    float* h    = ws + (size_t)N;
    float* hW   = ws + (size_t)17 * N;
    float* agg  = ws + (size_t)33 * N;

    const int T = 256;
    const int numTiles = (N + 15) / 16;
    const int gemmGrid = (numTiles + 7) / 8;
    const int nodeGrid = (N + T - 1) / T;
    const int epiGrid  = (N * 4 + T - 1) / T;
    const int zero4    = (N * 16) / 4;                  // N*16 divisible by 4
    const int zeroGrid = (zero4 + T - 1) / T;

    // Degree -> dinv  (deg over dst, +1 self-loop)
    gcn_fill_f32<<<nodeGrid, T, 0, stream>>>(dinv, 1.0f, N);
    gcn_deg_accum<<<(E + T - 1) / T, T, 0, stream>>>(dst, dinv, E);
    gcn_rsqrt<<<nodeGrid, T, 0, stream>>>(dinv, N);

    // Layer 1: 16 -> 8
    gcn_gemm_wmma<<<gemmGrid, T, 0, stream>>>(x, W1, hW, N, 16, 8);
    gcn_fill4_f32<<<zeroGrid, T, 0, stream>>>(agg, 0.0f, zero4);
    gcn_scatter<3><<<((E << 3) + T - 1) / T, T, 0, stream>>>(src, dst, dinv, hW, agg, E);
    gcn_epilogue_mish<<<epiGrid, T, 0, stream>>>(agg, hW, dinv, b1, h, N, 8);

    // Layer 2: 8 -> 16
    gcn_gemm_wmma<<<gemmGrid, T, 0, stream>>>(h, W2, hW, N, 8, 16);
    gcn_fill4_f32<<<zeroGrid, T, 0, stream>>>(agg, 0.0f, zero4);
    gcn_scatter<4><<<((E << 4) + T - 1) / T, T, 0, stream>>>(src, dst, dinv, hW, agg, E);
    gcn_epilogue_mish<<<epiGrid, T, 0, stream>>>(agg, hW, dinv, b2, h, N, 16);

    // Layer 3: 16 -> 8
    gcn_gemm_wmma<<<gemmGrid, T, 0, stream>>>(h, W3, hW, N, 16, 8);
    gcn_fill4_f32<<<zeroGrid, T, 0, stream>>>(agg, 0.0f, zero4);
    gcn_scatter<3><<<((E << 3) + T - 1) / T, T, 0, stream>>>(src, dst, dinv, hW, agg, E);
    gcn_epilogue_mish<<<epiGrid, T, 0, stream>>>(agg, hW, dinv, b3, h, N, 8);

    // Layer 4: 8 -> 2, then log_softmax
    gcn_gemm_wmma<<<gemmGrid, T, 0, stream>>>(h, W4, hW, N, 8, 2);
    gcn_fill4_f32<<<zeroGrid, T, 0, stream>>>(agg, 0.0f, zero4);
    gcn_scatter<1><<<((E << 1) + T - 1) / T, T, 0, stream>>>(src, dst, dinv, hW, agg, E);
    gcn_logsoftmax2<<<nodeGrid, T, 0, stream>>>(agg, hW, dinv, b4, out, N);
}